// LIFNeuron_3487513444972
// MI455X (gfx1250) — compile-verified
//
#include <hip/hip_runtime.h>

// LIF neuron scan: v = v*TAU + x + r; s = (v > 0); v = v * (1 - s)
// [T, B, N] = [100, 64, 2048], float32 in / float32 out.
// Memory-bound: 157 MB total traffic -> ~6.7 us floor at 23.3 TB/s HBM.
// Strategy: one thread owns 2 membrane chains (float2), loops over T.
// Depth-4 register software pipeline + global_prefetch_b8 (gfx1250 path)
// at a 16-timestep distance to keep the HBM/L2 pipe full.

#define LIF_TAU 0.5f

// ---------------------------------------------------------------------------
// Fast path: float2 vectorized, depth-4 pipelined. Requires BN even,
// T % 4 == 0, T >= 8.
// ---------------------------------------------------------------------------
__global__ __launch_bounds__(256) void lif_scan_v2(
    const float2* __restrict__ inp,   // [T, S] float2
    const float2* __restrict__ rec,   // [T, S] float2
    float2* __restrict__ out,         // [T, S] float2
    int T, int S)                     // S = (B*N)/2 in float2 units
{
    const int idx = blockIdx.x * blockDim.x + threadIdx.x;
    if (idx >= S) return;

    const float2* __restrict__ xp = inp + idx;
    const float2* __restrict__ rp = rec + idx;
    float2* __restrict__ op = out + idx;

    float vx = 0.0f, vy = 0.0f;

    constexpr int D  = 4;   // register pipeline depth (timesteps of loads in flight)
    constexpr int PF = 16;  // L2 prefetch distance in timesteps

    // Prologue: fill the pipeline with timesteps 0..D-1.
    float2 xb[D], rb[D];
#pragma unroll
    for (int i = 0; i < D; ++i) {
        xb[i] = xp[(size_t)i * S];
        rb[i] = rp[(size_t)i * S];
    }

    int t = 0;
    // Main loop: t in [0, T-D), step D. Loads run D steps ahead of compute.
    for (; t < T - D; t += D) {
        // Warm L2 well ahead of the demand stream (lowers to global_prefetch_b8).
        if (t + PF < T) {
            __builtin_prefetch(xp + (size_t)(t + PF) * S, 0, 3);
            __builtin_prefetch(rp + (size_t)(t + PF) * S, 0, 3);
        }
#pragma unroll
        for (int k = 0; k < D; ++k) {
            const float2 xv = xb[k];
            const float2 rv = rb[k];
            // Issue the replacement loads immediately (independent of compute).
            xb[k] = xp[(size_t)(t + k + D) * S];
            rb[k] = rp[(size_t)(t + k + D) * S];

            // Leak + integrate.
            vx = vx * LIF_TAU + xv.x + rv.x;
            vy = vy * LIF_TAU + xv.y + rv.y;
            // Fire.
            const float sx = (vx > 0.0f) ? 1.0f : 0.0f;
            const float sy = (vy > 0.0f) ? 1.0f : 0.0f;
            // Hard reset where spiked (select == v*(1-s) for s in {0,1}).
            vx = (vx > 0.0f) ? 0.0f : vx;
            vy = (vy > 0.0f) ? 0.0f : vy;

            op[(size_t)(t + k) * S] = make_float2(sx, sy);
        }
    }

    // Epilogue: final D timesteps, data already in the pipeline registers.
#pragma unroll
    for (int k = 0; k < D; ++k) {
        const float2 xv = xb[k];
        const float2 rv = rb[k];
        vx = vx * LIF_TAU + xv.x + rv.x;
        vy = vy * LIF_TAU + xv.y + rv.y;
        const float sx = (vx > 0.0f) ? 1.0f : 0.0f;
        const float sy = (vy > 0.0f) ? 1.0f : 0.0f;
        vx = (vx > 0.0f) ? 0.0f : vx;
        vy = (vy > 0.0f) ? 0.0f : vy;
        op[(size_t)(t + k) * S] = make_float2(sx, sy);
    }
}

// ---------------------------------------------------------------------------
// Generic fallback: scalar, any T / BN.
// ---------------------------------------------------------------------------
__global__ __launch_bounds__(256) void lif_scan_scalar(
    const float* __restrict__ inp,
    const float* __restrict__ rec,
    float* __restrict__ out,
    int T, long long BN)
{
    const long long idx = (long long)blockIdx.x * blockDim.x + threadIdx.x;
    if (idx >= BN) return;

    float v = 0.0f;
    for (int t = 0; t < T; ++t) {
        const size_t o = (size_t)t * BN + idx;
        if (t + 8 < T) __builtin_prefetch(inp + ((size_t)(t + 8) * BN + idx), 0, 3);
        v = v * LIF_TAU + inp[o] + rec[o];
        out[o] = (v > 0.0f) ? 1.0f : 0.0f;
        v = (v > 0.0f) ? 0.0f : v;
    }
}

extern "C" void kernel_launch(void* const* d_in, const int* in_sizes, int n_in,
                              void* d_out, int out_size, void* d_ws, size_t ws_size,
                              hipStream_t stream) {
    (void)n_in; (void)d_ws; (void)ws_size;

    const float* inp = (const float*)d_in[0];
    const float* rec = (const float*)d_in[1];
    float* out = (float*)d_out;

    const int T = 100;                                 // fixed by the reference
    const long long total = (long long)in_sizes[0];    // T * B * N
    const long long BN = total / T;                    // 64 * 2048 = 131072

    if ((BN % 2 == 0) && (T % 4 == 0) && (T >= 8)) {
        const int S = (int)(BN / 2);                   // 65536 float2 lanes
        const int threads = 256;                       // 8 wave32s / block
        const int blocks = (S + threads - 1) / threads;
        lif_scan_v2<<<blocks, threads, 0, stream>>>(
            (const float2*)inp, (const float2*)rec, (float2*)out, T, S);
    } else {
        const int threads = 256;
        const int blocks = (int)((BN + threads - 1) / threads);
        lif_scan_scalar<<<blocks, threads, 0, stream>>>(inp, rec, out, T, BN);
    }
}